// SimpleTransMatcherLoss_54228257079749
// MI455X (gfx1250) — compile-verified
//
#include <hip/hip_runtime.h>
#include <math.h>

typedef __attribute__((ext_vector_type(2))) float v2f;
typedef __attribute__((ext_vector_type(8))) float v8f;

#define BATCH    256
#define FEAT     98304            // 512*24*8
#define NT       (BATCH / 64)     // 4 tiles of 64 per side
#define NPAIR    (NT * (NT + 1) / 2)  // 10 upper-triangle 64x64 tiles
#define NKCHUNK  48               // K split: 10 * 48 = 480 blocks of 8 waves
#define KZ       (FEAT / NKCHUNK) // 2048, divisible by KB
#define KB       64               // K-slab staged in LDS per iteration
#define KSTRIDE  68               // 64+4 pad: bank = (4*row + k) % 64 -> conflict-free b64 reads

// ---------------------------------------------------------------- zero scores
__global__ void tm_zero_kernel(float* __restrict__ p, int n) {
    int i = blockIdx.x * blockDim.x + threadIdx.x;
    if (i < n) p[i] = 0.0f;
}

// ---------------------------------------------------------------- row inv-norms
__global__ __launch_bounds__(256)
void tm_norm_kernel(const float* __restrict__ A, float* __restrict__ inv_norm) {
    __shared__ float red[256];
    const int row = blockIdx.x;
    const int t   = threadIdx.x;
    const float* base = A + (size_t)row * FEAT;
    float s = 0.0f;
    // 24576 float4 per row / 256 threads = 96 float4 each, fully coalesced
    for (int i = 0; i < FEAT / 4; i += 256) {
        float4 v = *(const float4*)(base + (size_t)(i + t) * 4);
        s += v.x * v.x + v.y * v.y + v.z * v.z + v.w * v.w;
    }
    red[t] = s;
    __syncthreads();
    for (int off = 128; off > 0; off >>= 1) {
        if (t < off) red[t] += red[t + off];
        __syncthreads();
    }
    if (t == 0) {
        float n = sqrtf(red[0]);
        inv_norm[row] = 1.0f / fmaxf(n, 1e-8f);   // matches jnp.clip(norm, 1e-8)
    }
}

// ---------------------------------------------------------------- Gram via WMMA
// Block: 256 threads = 8 waves. Computes a 64x64 tile of the symmetric Gram
// matrix over a KZ-long K slice; only upper-triangle tiles are launched and
// off-diagonal results are mirrored. Each wave owns two 16x16 subtiles that
// share the A operand (2 WMMAs per A load). Diagonal tiles reuse the A slab
// for B, halving their L2 traffic.
__global__ __launch_bounds__(256)
void tm_gram_kernel(const float* __restrict__ A, float* __restrict__ S) {
    __shared__ float lds[2 * 64 * KSTRIDE];   // [0]: A slab, [64*KSTRIDE]: B slab

    const int t    = threadIdx.x;       // 0..255
    const int wave = t >> 5;            // 0..7
    const int lane = t & 31;

    // blockIdx.x -> (bi, bj) with bi <= bj over NT x NT 64-tiles
    int idx = blockIdx.x, bi = 0, span = NT;
    while (idx >= span) { idx -= span; ++bi; --span; }
    const int  bj   = bi + idx;
    const bool diag = (bi == bj);

    const int gi0 = bi * 64;            // output row base
    const int gj0 = bj * 64;            // output col base
    const int k0  = blockIdx.z * KZ;    // K slice base

    const int si = (wave >> 1) * 16;    // subtile row offset (0,16,32,48)
    const int sj = (wave & 1) * 32;     // first of two subtile cols (0,32)
    const int m   = lane & 15;          // row within operand tile
    const int sel = lane >> 4;          // K-half select (ISA fp32 A/B layout)

    const int boff = diag ? 0 : 64 * KSTRIDE;   // uniform LDS offset keeps ds_ path

    v8f acc0 = {}, acc1 = {};

    for (int kc = k0; kc < k0 + KZ; kc += KB) {
        // Cooperative fill: 1024 float4 per slab, 4 per thread, coalesced
        // (threads 0..15 cover one row's 64 floats = 256B contiguous).
#pragma unroll
        for (int p = 0; p < 4; ++p) {
            const int id  = t + p * 256;        // 0..1023
            const int row = id >> 4;            // 0..63
            const int c4  = (id & 15) << 2;     // 0,4,...,60
            *(float4*)(lds + row * KSTRIDE + c4) =
                *(const float4*)(A + (size_t)(gi0 + row) * FEAT + kc + c4);
            if (!diag)
                *(float4*)(lds + 64 * KSTRIDE + row * KSTRIDE + c4) =
                    *(const float4*)(A + (size_t)(gj0 + row) * FEAT + kc + c4);
        }
        // Prefetch next slab while this one computes.
        if (kc + KB < k0 + KZ) {
            const int row = t & 63;
            __builtin_prefetch(A + (size_t)(gi0 + row) * FEAT + kc + KB, 0, 1);
            if (!diag)
                __builtin_prefetch(A + (size_t)(gj0 + row) * FEAT + kc + KB, 0, 1);
        }
        __syncthreads();

        // 32 chained v_wmma_f32_16x16x4_f32 per slab per wave; A reused 2x.
#pragma unroll
        for (int kk = 0; kk < KB; kk += 4) {
            const int ko = kk + sel * 2;
            const v2f a2  = *(const v2f*)(lds + (si + m) * KSTRIDE + ko);
            const v2f b20 = *(const v2f*)(lds + boff + (sj + m) * KSTRIDE + ko);
            const v2f b21 = *(const v2f*)(lds + boff + (sj + 16 + m) * KSTRIDE + ko);
            acc0 = __builtin_amdgcn_wmma_f32_16x16x4_f32(
                false, a2, false, b20, (short)0, acc0, false, false);
            acc1 = __builtin_amdgcn_wmma_f32_16x16x4_f32(
                false, a2, false, b21, (short)0, acc1, false, false);
        }
        __syncthreads();
    }

    // D layout: VGPR v -> row (v + 8*sel), col m. Accumulate; mirror off-diag.
    const int col0 = gj0 + sj + m;
    const int col1 = col0 + 16;
#pragma unroll
    for (int v = 0; v < 8; ++v) {
        const int row = gi0 + si + v + sel * 8;
        atomicAdd(&S[row * BATCH + col0], acc0[v]);
        atomicAdd(&S[row * BATCH + col1], acc1[v]);
        if (!diag) {
            atomicAdd(&S[col0 * BATCH + row], acc0[v]);
            atomicAdd(&S[col1 * BATCH + row], acc1[v]);
        }
    }
}

// ---------------------------------------------------------------- mining + loss
__global__ __launch_bounds__(256)
void tm_loss_kernel(const float* __restrict__ S, const float* __restrict__ inv_norm,
                    const int* __restrict__ labels, float* __restrict__ out) {
    __shared__ float sh_inv[BATCH];
    __shared__ int   sh_lab[BATCH];
    __shared__ float r_trip[BATCH];
    __shared__ int   r_valid[BATCH];
    __shared__ int   r_corr[BATCH];

    const int i = threadIdx.x;
    sh_inv[i] = inv_norm[i];
    sh_lab[i] = labels[i];
    __syncthreads();

    const int   li = sh_lab[i];
    const float wi = sh_inv[i];
    float hp = 1e30f, hn = -1e30f, mp = -1e30f;
    bool has_pos = false, has_neg = false;
    for (int j = 0; j < BATCH; ++j) {
        const float s  = S[i * BATCH + j] * wi * sh_inv[j];
        const bool  eq = (sh_lab[j] == li);
        if (eq) {
            if (j != i) {
                hp = fminf(hp, s);
                mp = fmaxf(mp, s);
                has_pos = true;
            }
        } else {
            hn = fmaxf(hn, s);
            has_neg = true;
        }
    }
    const bool  valid = has_pos && has_neg;
    const float trip  = fmaxf(hn - hp + 1.0f, 0.0f);
    r_trip[i]  = valid ? trip : 0.0f;
    r_valid[i] = valid ? 1 : 0;
    r_corr[i]  = (valid && (mp > hn)) ? 1 : 0;
    __syncthreads();
    for (int off = 128; off > 0; off >>= 1) {
        if (i < off) {
            r_trip[i]  += r_trip[i + off];
            r_valid[i] += r_valid[i + off];
            r_corr[i]  += r_corr[i + off];
        }
        __syncthreads();
    }
    if (i == 0) {
        const int   nv    = r_valid[0];
        const float denom = (float)(nv > 1 ? nv : 1);
        out[0] = (nv > 0) ? (r_trip[0] / denom) : 0.0f;          // loss
        out[1] = (nv > 0) ? ((float)r_corr[0] / denom) : 0.5f;   // accuracy
    }
}

// ---------------------------------------------------------------- launch
extern "C" void kernel_launch(void* const* d_in, const int* in_sizes, int n_in,
                              void* d_out, int out_size, void* d_ws, size_t ws_size,
                              hipStream_t stream) {
    const float* feats  = (const float*)d_in[0];   // [256, 512, 24, 8] fp32
    const int*   labels = (const int*)d_in[1];     // [256] int32
    float*       out    = (float*)d_out;           // (loss, accuracy)

    float* inv_norm = (float*)d_ws;                // 256 floats
    float* scores   = inv_norm + BATCH;            // 256*256 floats

    tm_zero_kernel<<<(BATCH * BATCH + 255) / 256, 256, 0, stream>>>(scores, BATCH * BATCH);
    tm_norm_kernel<<<BATCH, 256, 0, stream>>>(feats, inv_norm);

    dim3 grid(NPAIR, 1, NKCHUNK);                  // 10 x 48 = 480 blocks
    tm_gram_kernel<<<grid, 256, 0, stream>>>(feats, scores);

    tm_loss_kernel<<<1, BATCH, 0, stream>>>(scores, inv_norm, labels, out);
}